// TensorAugment_79216376807666
// MI455X (gfx1250) — compile-verified
//
#include <hip/hip_runtime.h>
#include <hip/hip_bf16.h>

typedef __attribute__((ext_vector_type(2))) float v2f;
typedef __attribute__((ext_vector_type(8))) float v8f;

#define HH 4096
#define WW 4096
#define IMH 2048
#define IMW 2048

// E tile: 40 rows (i, with 4-halo) x 72 cols (j, with 4-halo)
#define EROWS 40
#define ECOLS 72
// per-wave V tile: 16 rows x 24 cols, stride 25 (bank-conflict-free: gcd(25,64)=1)
#define VSTRIDE 25

__global__ __launch_bounds__(256) void fused_aug_blur(
    const float* __restrict__ img,
    const float* __restrict__ noise,
    float* __restrict__ out)
{
    __shared__ float Etile[EROWS * ECOLS];
    __shared__ float Vtile[8 * 16 * VSTRIDE];
    __shared__ float ktabz[48];   // zero-padded band table: ktabz[16+d] = k[d] for d in [0,8]

    const int tid = threadIdx.x;
    const int i0 = blockIdx.y * 32;   // output row base of workgroup
    const int j0 = blockIdx.x * 64;   // output col base of workgroup

    // ---- build Gaussian band table (f32, matches numpy float32 construction) ----
    if (tid < 48) {
        float v = 0.0f;
        int d = tid - 16;
        if (d >= 0 && d <= 8) {
            float s = 0.0f, w = 0.0f;
            #pragma unroll
            for (int u = 0; u < 9; ++u) {
                float t = (float)(u - 4);
                float e = expf(-0.5f * t * t);
                s += e;
                if (u == d) w = e;
            }
            v = w / s;
        }
        ktabz[tid] = v;
    }

    // ---- phase 1: fill E tile ----
    // E[i,j] = Up[4095-j, i] + 0.01*noise[i,j]; symmetric reflection at borders.
    for (int e = tid; e < EROWS * ECOLS; e += 256) {
        int ei = e / ECOLS;
        int ej = e - ei * ECOLS;
        int gi = i0 - 4 + ei;
        int gj = j0 - 4 + ej;
        gi = (gi < 0) ? (-1 - gi) : ((gi >= HH) ? (2 * HH - 1 - gi) : gi);
        gj = (gj < 0) ? (-1 - gj) : ((gj >= WW) ? (2 * WW - 1 - gj) : gj);

        int y = (HH - 1) - gj;    // row in upsampled image
        int x = gi;               // col in upsampled image

        // bilinear, half-pixel centers: coord = y*0.5 - 0.25
        int yh = y >> 1;
        int r0, r1; float wr0, wr1;
        if ((y & 1) == 0) { r0 = yh - 1; r1 = yh;     wr0 = 0.25f; wr1 = 0.75f; }
        else              { r0 = yh;     r1 = yh + 1; wr0 = 0.75f; wr1 = 0.25f; }
        r0 = max(r0, 0); r1 = min(r1, IMH - 1);
        int xh = x >> 1;
        int c0, c1; float wc0, wc1;
        if ((x & 1) == 0) { c0 = xh - 1; c1 = xh;     wc0 = 0.25f; wc1 = 0.75f; }
        else              { c0 = xh;     c1 = xh + 1; wc0 = 0.75f; wc1 = 0.25f; }
        c0 = max(c0, 0); c1 = min(c1, IMW - 1);

        const float* rp0 = img + (size_t)r0 * IMW;
        const float* rp1 = img + (size_t)r1 * IMW;
        float a  = wc0 * rp0[c0] + wc1 * rp0[c1];
        float b  = wc0 * rp1[c0] + wc1 * rp1[c1];
        float up = wr0 * a + wr1 * b;
        float nz = noise[(size_t)gi * WW + gj];
        Etile[e] = up + 0.01f * nz;
    }
    __syncthreads();

    // ---- wave geometry: 8 waves -> 2 (i) x 4 (j) grid of 16x16 output tiles ----
    const int lane = tid & 31;
    const int wave = tid >> 5;
    const int wy = wave >> 2;         // 0..1
    const int wx = wave & 3;          // 0..3
    const int h  = lane >> 4;         // half-wave
    const int n  = lane & 15;         // lane-in-half
    const int erow0 = wy * 16;        // wave's E window rows [erow0, erow0+23]
    const int ecol0 = wx * 16;        // wave's E window cols [ecol0, ecol0+23]

    // ---- vertical blur via WMMA: V(16x24) = KV(16x24) * E(24x24) ----
    v8f acc0 = {};  // cols [0,15] of V
    v8f acc1 = {};  // cols [8,23] of V
    #pragma unroll
    for (int kk = 0; kk < 6; ++kk) {
        const int t0 = kk * 4 + 2 * h;
        const int t1 = t0 + 1;
        v2f a;                                   // A = KV, lane row m = n
        a.x = ktabz[16 + t0 - n];
        a.y = ktabz[16 + t1 - n];
        v2f b0, b1;                              // B = E rows t0/t1
        b0.x = Etile[(erow0 + t0) * ECOLS + ecol0 + 0 + n];
        b0.y = Etile[(erow0 + t1) * ECOLS + ecol0 + 0 + n];
        b1.x = Etile[(erow0 + t0) * ECOLS + ecol0 + 8 + n];
        b1.y = Etile[(erow0 + t1) * ECOLS + ecol0 + 8 + n];
        acc0 = __builtin_amdgcn_wmma_f32_16x16x4_f32(false, a, false, b0, (short)0, acc0, false, false);
        acc1 = __builtin_amdgcn_wmma_f32_16x16x4_f32(false, a, false, b1, (short)0, acc1, false, false);
    }

    // ---- transpose V through LDS (per-wave private region) ----
    float* V = &Vtile[wave * 16 * VSTRIDE];
    #pragma unroll
    for (int v = 0; v < 8; ++v) {
        V[(v + 8 * h) * VSTRIDE + 0 + n] = acc0[v];
        V[(v + 8 * h) * VSTRIDE + 8 + n] = acc1[v];
    }
    __syncthreads();

    // ---- horizontal blur via WMMA: out(16x16) = V(16x24) * KH(24x16) ----
    v8f acc = {};
    #pragma unroll
    for (int kk = 0; kk < 6; ++kk) {
        const int t0 = kk * 4 + 2 * h;
        const int t1 = t0 + 1;
        v2f a;                                   // A = V, lane row m = n
        a.x = V[n * VSTRIDE + t0];
        a.y = V[n * VSTRIDE + t1];
        v2f b;                                   // B = KH band constants
        b.x = ktabz[16 + t0 - n];
        b.y = ktabz[16 + t1 - n];
        acc = __builtin_amdgcn_wmma_f32_16x16x4_f32(false, a, false, b, (short)0, acc, false, false);
    }

    // ---- epilogue: log1p(relu(.)) and coalesced store ----
    const int orow0 = i0 + wy * 16;
    const int ocol0 = j0 + wx * 16;
    #pragma unroll
    for (int v = 0; v < 8; ++v) {
        float val = acc[v];
        val = fmaxf(val, 0.0f);
        val = log1pf(val);
        out[(size_t)(orow0 + v + 8 * h) * WW + (ocol0 + n)] = val;
    }
}

extern "C" void kernel_launch(void* const* d_in, const int* in_sizes, int n_in,
                              void* d_out, int out_size, void* d_ws, size_t ws_size,
                              hipStream_t stream) {
    const float* img   = (const float*)d_in[0];   // (1, 2048, 2048) f32
    const float* noise = (const float*)d_in[1];   // (1, 4096, 4096) f32
    float* out = (float*)d_out;                   // (1, 4096, 4096) f32

    dim3 grid(WW / 64, HH / 32, 1);   // 64 x 128 workgroups
    dim3 block(256, 1, 1);
    fused_aug_blur<<<grid, block, 0, stream>>>(img, noise, out);
}